// HLoss2_5428838662496
// MI455X (gfx1250) — compile-verified
//
#include <hip/hip_runtime.h>
#include <math.h>

// ---------------------------------------------------------------------------
// HLoss entropy kernel for gfx1250 (MI455X).
// Heavy phase: 300 x 1e6 Gaussian kernel matrix. Exponent arguments computed
// as a K=4 GEMM via V_WMMA_F32_16X16X4_F32 (1 instr per 256 (m,n) pairs),
// exponentials via v_exp_f32 (log2e folded into the quadratic coefficients).
// ---------------------------------------------------------------------------

typedef float v2f __attribute__((ext_vector_type(2)));
typedef float v8f __attribute__((ext_vector_type(8)));

#define LOG2E        1.4426950408889634f
#define LN2          0.6931471805599453f
#define SIGMA        0.1f
#define INV_SQRT2PI  0.3989422804014327f
// c = (-0.5 / sigma^2) * log2(e)  -> exponent argument directly usable by v_exp_f32
#define C_EXP        (-50.0f * LOG2E)

#define NB1 1024   // blocks for stats pass
#define GX  96     // n-direction blocks for KDE pass
#define WPB 8      // waves per 256-thread block (wave32)

// ---------------- pass 1: sum(r), sum(r^2) in double --------------------------
__global__ __launch_bounds__(256) void k_stats(const float* __restrict__ x1,
                                               const float* __restrict__ x2,
                                               int N, double* __restrict__ part) {
  __shared__ double sh_s[256];
  __shared__ double sh_q[256];
  const int tid = threadIdx.x;
  double s = 0.0, q = 0.0;
  for (int i = blockIdx.x * blockDim.x + tid; i < N; i += gridDim.x * blockDim.x) {
    float r = x1[i] - x2[i];
    s += (double)r;
    q += (double)r * (double)r;
  }
  sh_s[tid] = s; sh_q[tid] = q;
  __syncthreads();
  for (int o = 128; o > 0; o >>= 1) {
    if (tid < o) { sh_s[tid] += sh_s[tid + o]; sh_q[tid] += sh_q[tid + o]; }
    __syncthreads();
  }
  if (tid == 0) { part[2 * blockIdx.x] = sh_s[0]; part[2 * blockIdx.x + 1] = sh_q[0]; }
}

// ---------------- pass 2: mu/std, samples, likelihoods, A-coefficients --------
__global__ __launch_bounds__(256) void k_setup(const double* __restrict__ part,
                                               const float* __restrict__ z,
                                               int N, int M, int MT16,
                                               float* __restrict__ a0,
                                               float* __restrict__ a1,
                                               float* __restrict__ samp,
                                               float* __restrict__ lik) {
  __shared__ double sh_s[256];
  __shared__ double sh_q[256];
  __shared__ float sh_mu, sh_sd;
  const int tid = threadIdx.x;
  double s = 0.0, q = 0.0;
  for (int i = tid; i < NB1; i += 256) { s += part[2 * i]; q += part[2 * i + 1]; }
  sh_s[tid] = s; sh_q[tid] = q;
  __syncthreads();
  for (int o = 128; o > 0; o >>= 1) {
    if (tid < o) { sh_s[tid] += sh_s[tid + o]; sh_q[tid] += sh_q[tid + o]; }
    __syncthreads();
  }
  if (tid == 0) {
    double S = sh_s[0], Q = sh_q[0];
    double mu  = S / (double)N;
    double var = (Q - S * S / (double)N) / (double)(N - 1);  // ddof=1
    sh_mu = (float)mu;
    sh_sd = (float)sqrt(var);
  }
  __syncthreads();
  const float mu = sh_mu, sd = sh_sd;
  for (int m = tid; m < MT16; m += 256) {
    float zm = (m < M) ? z[m] : 0.0f;          // padded rows: results ignored
    float sm = zm * sd + mu;                   // sample
    samp[m] = sm;
    a0[m] = C_EXP * sm * sm;                   // K=0 coefficient
    a1[m] = -2.0f * C_EXP * sm;                // K=1 coefficient
    // normal(sample, mu, sd): (sample-mu)/sd == z[m] exactly
    lik[m] = __builtin_amdgcn_exp2f(-0.5f * LOG2E * zm * zm) * (INV_SQRT2PI / sd);
  }
}

// ---------------- pass 3: KDE kernel matrix via WMMA --------------------------
// grid = (GX, MT). Each wave owns one 16-m tile and strided 16-n chunks.
__global__ __launch_bounds__(256) void k_kde(const float* __restrict__ x1,
                                             const float* __restrict__ x2,
                                             const float* __restrict__ a0,
                                             const float* __restrict__ a1,
                                             int nFull, float* __restrict__ P) {
  const int lane = threadIdx.x & 31;
  const int wave = threadIdx.x >> 5;
  const int waveGlobal = blockIdx.x * WPB + wave;
  const int totalWaves = gridDim.x * WPB;
  const int mb = blockIdx.y * 16;
  const bool lo = lane < 16;       // lanes 0-15: K=0,1 ; lanes 16-31: K=2,3
  const int  nl = lane & 15;

  // A matrix (16x4, f32, 2 VGPRs): per-m coefficients in low half,
  // row-invariant constants (c, 0) in high half (K=2,3). No EXEC divergence.
  float t0 = a0[mb + nl];
  float t1 = a1[mb + nl];
  v2f A;
  A[0] = lo ? t0 : C_EXP;
  A[1] = lo ? t1 : 0.0f;

  v8f acc = {0.f, 0.f, 0.f, 0.f, 0.f, 0.f, 0.f, 0.f};

  for (int nb = waveGlobal * 16; nb < nFull; nb += totalWaves * 16) {
    const int n = nb + nl;
    float r = x1[n] - x2[n];
    // B matrix (4x16, f32, 2 VGPRs): rows [1, r, r^2, 0]
    v2f B;
    B[0] = lo ? 1.0f : r * r;     // K=0 row | K=2 row
    B[1] = lo ? r    : 0.0f;      // K=1 row | K=3 row
    v8f zero = {0.f, 0.f, 0.f, 0.f, 0.f, 0.f, 0.f, 0.f};
    // D[m][n] = c*(s_m - r_n)^2 * log2e  for the whole 16x16 tile
    v8f d = __builtin_amdgcn_wmma_f32_16x16x4_f32(
        /*neg_a=*/false, A, /*neg_b=*/false, B,
        /*c_mod=*/(short)0, zero, /*reuse_a=*/false, /*reuse_b=*/false);
#pragma unroll
    for (int i = 0; i < 8; ++i) acc[i] += __builtin_amdgcn_exp2f(d[i]);
  }

  // Reduce each acc[i] across the 16 n-lanes of each half-wave (halves stay
  // separate: xor touches only bits 0..3).
#pragma unroll
  for (int i = 0; i < 8; ++i) {
    float v = acc[i];
    v += __shfl_xor(v, 1, 32);
    v += __shfl_xor(v, 2, 32);
    v += __shfl_xor(v, 4, 32);
    v += __shfl_xor(v, 8, 32);
    acc[i] = v;
  }

  // Stage partials: P[tile][waveGlobal][16]. D layout: VGPR i -> m=i (lanes
  // 0-15) and m=8+i (lanes 16-31).
  float* dst = P + ((size_t)blockIdx.y * totalWaves + waveGlobal) * 16;
  if (lane == 0) {
#pragma unroll
    for (int i = 0; i < 8; ++i) dst[i] = acc[i];
  } else if (lane == 16) {
#pragma unroll
    for (int i = 0; i < 8; ++i) dst[8 + i] = acc[i];
  }
}

// ---------------- pass 4: finalize ------------------------------------------
__global__ __launch_bounds__(512) void k_final(const float* __restrict__ P,
                                               const float* __restrict__ samp,
                                               const float* __restrict__ lik,
                                               const float* __restrict__ x1,
                                               const float* __restrict__ x2,
                                               int N, int M, int WV,
                                               float* __restrict__ out) {
  __shared__ float sh[512];
  const int tid = threadIdx.x;
  const int nFull = N & ~15;
  float local = 0.0f;
  for (int m = tid; m < M; m += blockDim.x) {
    const int tile = m >> 4;
    const int j = m & 15;
    const float* base = P + ((size_t)tile * WV) * 16 + j;
    float s = 0.0f;
    for (int w = 0; w < WV; ++w) s += base[(size_t)w * 16];
    // scalar tail for N % 16 (zero for N = 1e6)
    float sm = samp[m];
    for (int n = nFull; n < N; ++n) {
      float r = x1[n] - x2[n];
      float dd = sm - r;
      s += __builtin_amdgcn_exp2f(C_EXP * dd * dd);
    }
    float p = s * (INV_SQRT2PI / SIGMA) / (float)N;          // mean of pdf
    float lg = __builtin_amdgcn_logf(p + 1e-8f) * LN2;       // natural log
    local += (-p * lg) / lik[m];
  }
  sh[tid] = local;
  __syncthreads();
  for (int o = 256; o > 0; o >>= 1) {
    if (tid < o) sh[tid] += sh[tid + o];
    __syncthreads();
  }
  if (tid == 0) out[0] = sh[0] / (float)M;
}

// ---------------------------------------------------------------------------
extern "C" void kernel_launch(void* const* d_in, const int* in_sizes, int n_in,
                              void* d_out, int out_size, void* d_ws, size_t ws_size,
                              hipStream_t stream) {
  const float* x1 = (const float*)d_in[0];
  const float* x2 = (const float*)d_in[1];
  const float* z  = (const float*)d_in[2];
  const int N = in_sizes[0];
  const int M = in_sizes[2];
  const int MT = (M + 15) / 16;      // 16-row m-tiles (19 for M=300)
  const int MT16 = MT * 16;
  const int WV = GX * WPB;           // waves per m-tile
  const int nFull = N & ~15;

  // workspace layout (~0.95 MB for N=1e6, M=300)
  char* ws = (char*)d_ws;
  double* part = (double*)ws;                                   // NB1*2 doubles
  float* a0   = (float*)(ws + (size_t)NB1 * 2 * sizeof(double));
  float* a1   = a0 + MT16;
  float* samp = a1 + MT16;
  float* lik  = samp + MT16;
  float* P    = lik + MT16;                                     // MT*WV*16 floats

  k_stats<<<NB1, 256, 0, stream>>>(x1, x2, N, part);
  k_setup<<<1, 256, 0, stream>>>(part, z, N, M, MT16, a0, a1, samp, lik);
  dim3 g2(GX, MT);
  k_kde<<<g2, 256, 0, stream>>>(x1, x2, a0, a1, nFull, P);
  k_final<<<1, 512, 0, stream>>>(P, samp, lik, x1, x2, N, M, WV, (float*)d_out);
}